// Model_46239617909135
// MI455X (gfx1250) — compile-verified
//
#include <hip/hip_runtime.h>
#include <math.h>

typedef float v2f __attribute__((ext_vector_type(2)));
typedef float v8f __attribute__((ext_vector_type(8)));

namespace {
constexpr int B_TOT = 2048;
constexpr int T_LEN = 66;
constexpr int INW   = 9;
constexpr int H     = 100;
constexpr int NMEM  = 128;
constexpr int D     = 20;
constexpr int OUTW  = 8;
constexpr int NH    = 2;
constexpr int PPH   = 26;      // D+1+1+3+1
constexpr int PDIM  = 92;      // NH*PPH + 2*D
constexpr int PPAD  = 96;      // padded to 6 WMMA col tiles
constexpr int G4    = 400;     // 4*H
constexpr int KZ    = 132;     // 129 (x|r|h) padded to mult of 4
constexpr int BC    = 16;      // batch rows per workgroup == WMMA M
constexpr int NT    = 256;     // 8 waves
constexpr float CLIPV = 20.f;
constexpr float EPSV  = 1e-8f;

// packed-weight workspace layout (floats)
constexpr int WS_WCAT = 0;                 // [KZ][G4]  (Wx;Wh;0-pad)
constexpr int WS_WP   = WS_WCAT + KZ*G4;   // [H][PPAD] (W_params 0-padded cols)
constexpr int WS_BP   = WS_WP   + H*PPAD;  // [PPAD]
constexpr int WS_TOT  = WS_BP   + PPAD;

constexpr int SMEM_FLOATS =
    BC*NMEM*D          // sM
  + BC*NH*NMEM         // sW   (w_prev / w)
  + BC*NH*NMEM         // sA   (beta*cos -> w_g)
  + BC*NH*NMEM         // sBuf (w_t^gamma)
  + BC*H               // sH
  + BC*H               // sC
  + BC*D               // sR
  + BC*KZ              // sZ
  + BC*G4              // sG (gates)
  + BC*PPAD            // sP
  + BC*NH*D            // sK
  + 32                 // sKn
  + 32                 // sBe
  + 32                 // sGg
  + 32                 // sGa
  + 32*3               // sS (shift weights)
  + 64                 // sRed (max,sum per bh)
  + BC*D               // sE (erase)
  + BC*D;              // sAd (add)
} // namespace

__device__ __forceinline__ float sigm(float x) { return 1.f / (1.f + expf(-x)); }
__device__ __forceinline__ float softplusf(float x) { return (x > 20.f) ? x : log1pf(expf(x)); }
__device__ __forceinline__ float clipf(float x) { return fminf(fmaxf(x, -CLIPV), CLIPV); }

// ---- prologue: pack weights into branch-free, zero-padded layouts ----
__global__ __launch_bounds__(256) void pack_weights(
    const float* __restrict__ Wx, const float* __restrict__ Wh,
    const float* __restrict__ W_params, const float* __restrict__ b_params,
    float* __restrict__ ws)
{
  int i = blockIdx.x * 256 + threadIdx.x;
  if (i < KZ*G4) {
    int k = i / G4, n = i - k*G4;
    float v = (k < 29) ? Wx[k*G4 + n] : ((k < 129) ? Wh[(k - 29)*G4 + n] : 0.f);
    ws[WS_WCAT + i] = v;
  } else if (i < KZ*G4 + H*PPAD) {
    int j = i - KZ*G4;
    int k = j / PPAD, n = j - k*PPAD;
    ws[WS_WP + j] = (n < PDIM) ? W_params[k*PDIM + n] : 0.f;
  } else if (i < WS_TOT) {
    int n = i - KZ*G4 - H*PPAD;
    ws[WS_BP + n] = (n < PDIM) ? b_params[n] : 0.f;
  }
}

__global__ __launch_bounds__(NT) void ntm_fused(
    const float* __restrict__ inputs,
    const float* __restrict__ Wcat,      // [KZ][G4]
    const float* __restrict__ b_lstm,
    const float* __restrict__ Wp,        // [H][PPAD]
    const float* __restrict__ bp,        // [PPAD]
    const float* __restrict__ W_out,
    const float* __restrict__ b_out,
    const int*   __restrict__ seq_ptr,
    float* __restrict__ out)
{
  extern __shared__ float smem[];
  float* sM   = smem;                      // [BC][NMEM][D]
  float* sW   = sM   + BC*NMEM*D;          // [BC*NH][NMEM]
  float* sA   = sW   + BC*NH*NMEM;         // scratch: beta*cos -> w_g
  float* sBuf = sA   + BC*NH*NMEM;         // scratch: sharpened pow
  float* sH   = sBuf + BC*NH*NMEM;         // [BC][H]
  float* sC   = sH   + BC*H;               // [BC][H]
  float* sR   = sC   + BC*H;               // [BC][D]
  float* sZ   = sR   + BC*D;               // [BC][KZ]
  float* sG   = sZ   + BC*KZ;              // [BC][G4]
  float* sP   = sG   + BC*G4;              // [BC][PPAD]
  float* sK   = sP   + BC*PPAD;            // [BC*NH][D]
  float* sKn  = sK   + BC*NH*D;            // [32]
  float* sBe  = sKn  + 32;                 // [32]
  float* sGg  = sBe  + 32;                 // [32]
  float* sGa  = sGg  + 32;                 // [32]
  float* sS   = sGa  + 32;                 // [32][3]
  float* sRed = sS   + 32*3;               // [32][2]
  float* sE   = sRed + 64;                 // [BC][D]
  float* sAd  = sE   + BC*D;               // [BC][D]

  const int tid   = threadIdx.x;
  const int wave  = tid >> 5;
  const int lane  = tid & 31;
  const int m     = lane & 15;                 // WMMA M row (batch)
  const int khalf = (lane < 16) ? 0 : 2;       // A/B fragment K split
  const int crow  = (lane < 16) ? 0 : 8;       // C/D row split
  const int b0    = blockIdx.x * BC;
  const int seq   = seq_ptr[0];
  const int tOutStart = seq + 1;
  const int TOUT  = T_LEN - tOutStart;

  // ---- initial recurrent state ----
  for (int i = tid; i < BC*NMEM*D;  i += NT) sM[i] = 1e-6f;
  for (int i = tid; i < BC*NH*NMEM; i += NT) sW[i] = ((i & (NMEM-1)) == 0) ? 1.f : 0.f;
  for (int i = tid; i < BC*H;       i += NT) { sH[i] = 0.f; sC[i] = 0.f; }
  for (int i = tid; i < BC*D;       i += NT) sR[i] = 0.f;
  __syncthreads();

  for (int t = 0; t < T_LEN; ++t) {
    // ---- phase 1: z = [x_t | r | h | 0] ----
    for (int i = tid; i < BC*KZ; i += NT) {
      int b = i / KZ, j = i - b*KZ;
      float v;
      if (j < INW)          v = inputs[(size_t)(b0 + b) * T_LEN * INW + t * INW + j];
      else if (j < INW + D) v = sR[b*D + (j - INW)];
      else if (j < 129)     v = sH[b*H + (j - 29)];
      else                  v = 0.f;
      sZ[i] = v;
    }
    __syncthreads();

    // ---- phase 2: gates = z @ Wcat + b  (fp32 WMMA, M=16 K=132 N=400) ----
    for (int ct = wave; ct < G4/16; ct += NT/32) {
      int n = ct*16 + m;
      const float* __restrict__ wc = Wcat + n;    // column base, stride G4
      float bias = b_lstm[n];
      v8f acc = { bias, bias, bias, bias, bias, bias, bias, bias };
      for (int kt = 0; kt < KZ/4; ++kt) {
        int k0 = kt*4 + khalf;
        v2f av; av.x = sZ[m*KZ + k0];     av.y = sZ[m*KZ + k0 + 1];
        v2f bv; bv.x = wc[(size_t)k0*G4]; bv.y = wc[(size_t)(k0 + 1)*G4];
        acc = __builtin_amdgcn_wmma_f32_16x16x4_f32(false, av, false, bv,
                                                    (short)0, acc, false, false);
      }
      #pragma unroll
      for (int v = 0; v < 8; ++v) sG[(v + crow)*G4 + n] = acc[v];
    }
    __syncthreads();

    // ---- phase 3: LSTM cell ----
    for (int i = tid; i < BC*H; i += NT) {
      int b = i / H, u = i - b*H;
      float gi = sG[b*G4 + u];
      float gf = sG[b*G4 + H   + u];
      float gg = sG[b*G4 + 2*H + u];
      float go = sG[b*G4 + 3*H + u];
      float c  = sigm(gf) * sC[i] + sigm(gi) * tanhf(gg);
      float h  = sigm(go) * tanhf(c);
      sC[i] = c; sH[i] = h;
    }
    __syncthreads();

    // ---- phase 4: p = clip(h @ Wp + bp)  (fp32 WMMA, K=100 N=96, branch-free) ----
    for (int ct = wave; ct < PPAD/16; ct += NT/32) {
      int n = ct*16 + m;
      const float* __restrict__ wp = Wp + n;      // column base, stride PPAD
      float bias = bp[n];
      v8f acc = { bias, bias, bias, bias, bias, bias, bias, bias };
      for (int kt = 0; kt < H/4; ++kt) {
        int k0 = kt*4 + khalf;
        v2f av; av.x = sH[m*H + k0];        av.y = sH[m*H + k0 + 1];
        v2f bv; bv.x = wp[(size_t)k0*PPAD]; bv.y = wp[(size_t)(k0 + 1)*PPAD];
        acc = __builtin_amdgcn_wmma_f32_16x16x4_f32(false, av, false, bv,
                                                    (short)0, acc, false, false);
      }
      #pragma unroll
      for (int v = 0; v < 8; ++v) sP[(v + crow)*PPAD + n] = clipf(acc[v]);
    }
    __syncthreads();

    // ---- phase 5: per-(b,head) addressing params ----
    if (tid < BC*NH) {
      int b = tid >> 1, hd = tid & 1;
      const float* pb = sP + b*PPAD + hd*PPH;
      float kn2 = 0.f;
      for (int d = 0; d < D; ++d) {
        float kv = tanhf(pb[d]);
        sK[tid*D + d] = kv;
        kn2 += kv * kv;
      }
      sKn[tid] = sqrtf(kn2);
      sBe[tid] = softplusf(pb[D]);
      sGg[tid] = sigm(pb[D + 1]);
      float s0 = pb[D + 2], s1 = pb[D + 3], s2 = pb[D + 4];
      float mx = fmaxf(s0, fmaxf(s1, s2));
      float e0 = expf(s0 - mx), e1 = expf(s1 - mx), e2 = expf(s2 - mx);
      float inv = 1.f / (e0 + e1 + e2);
      sS[tid*3 + 0] = e0 * inv; sS[tid*3 + 1] = e1 * inv; sS[tid*3 + 2] = e2 * inv;
      sGa[tid] = softplusf(pb[D + 5]) + 1.f;
    }
    __syncthreads();

    // ---- phase 6: beta * cosine(k, M) ----
    for (int i = tid; i < BC*NMEM; i += NT) {
      int b = i >> 7, n = i & (NMEM - 1);
      const float* Mr = sM + (b*NMEM + n)*D;
      const float* k0 = sK + (b*NH + 0)*D;
      const float* k1 = sK + (b*NH + 1)*D;
      float mn2 = 0.f, d0 = 0.f, d1 = 0.f;
      for (int d = 0; d < D; ++d) {
        float mv = Mr[d];
        mn2 += mv * mv; d0 += k0[d] * mv; d1 += k1[d] * mv;
      }
      float mn = sqrtf(mn2);
      sA[(b*NH + 0)*NMEM + n] = sBe[b*NH + 0] * d0 / (sKn[b*NH + 0]*mn + EPSV);
      sA[(b*NH + 1)*NMEM + n] = sBe[b*NH + 1] * d1 / (sKn[b*NH + 1]*mn + EPSV);
    }
    __syncthreads();

    // ---- phase 7: softmax reduction (max, sum-exp) per (b,head) ----
    if (tid < BC*NH) {
      const float* row = sA + tid*NMEM;
      float mx = row[0];
      for (int n = 1; n < NMEM; ++n) mx = fmaxf(mx, row[n]);
      float sum = 0.f;
      for (int n = 0; n < NMEM; ++n) sum += expf(row[n] - mx);
      sRed[tid*2 + 0] = mx;
      sRed[tid*2 + 1] = 1.f / sum;
    }
    __syncthreads();

    // ---- phase 8: w_c -> interpolate with w_prev -> w_g (in place) ----
    for (int i = tid; i < BC*NH*NMEM; i += NT) {
      int bh = i >> 7;
      float wc = expf(sA[i] - sRed[bh*2]) * sRed[bh*2 + 1];
      float g  = sGg[bh];
      sA[i] = g * wc + (1.f - g) * sW[i];
    }
    __syncthreads();

    // ---- phase 9: circular shift + sharpen pow ----
    for (int i = tid; i < BC*NH*NMEM; i += NT) {
      int bh = i >> 7, n = i & (NMEM - 1);
      const float* wg = sA + bh*NMEM;
      float wt = sS[bh*3 + 0] * wg[(n + 1) & (NMEM - 1)]
               + sS[bh*3 + 1] * wg[n]
               + sS[bh*3 + 2] * wg[(n - 1) & (NMEM - 1)];
      wt = fmaxf(wt, 1e-16f);
      sBuf[i] = expf(sGa[bh] * logf(wt));        // wt^gamma
    }
    __syncthreads();

    // ---- phase 10: pow-sum ----
    if (tid < BC*NH) {
      const float* row = sBuf + tid*NMEM;
      float sum = 0.f;
      for (int n = 0; n < NMEM; ++n) sum += row[n];
      sRed[tid*2] = 1.f / sum;
    }
    __syncthreads();

    // ---- phase 11: normalize -> new w (becomes w_prev) ----
    for (int i = tid; i < BC*NH*NMEM; i += NT)
      sW[i] = sBuf[i] * sRed[(i >> 7)*2];
    __syncthreads();

    // ---- phase 12: read vector r_new; erase/add vectors ----
    for (int i = tid; i < BC*D; i += NT) {
      int b = i / D, d = i - b*D;
      const float* wr = sW + (b*NH + 0)*NMEM;
      float acc = 0.f;
      for (int n = 0; n < NMEM; ++n) acc += wr[n] * sM[(b*NMEM + n)*D + d];
      sR[i]  = acc;
      sE[i]  = sigm (sP[b*PPAD + NH*PPH + d]);
      sAd[i] = tanhf(sP[b*PPAD + NH*PPH + D + d]);
    }
    __syncthreads();

    // ---- phase 13: memory erase + add ----
    for (int i = tid; i < BC*NMEM*D; i += NT) {
      int b   = i / (NMEM*D);
      int rem = i - b*NMEM*D;
      int n   = rem / D, d = rem - n*D;
      float ww = sW[(b*NH + 1)*NMEM + n];
      float mv = sM[i];
      sM[i] = mv * (1.f - ww * sE[b*D + d]) + ww * sAd[b*D + d];
    }

    // ---- phase 14: output projection + sigmoid (only t >= seq+1) ----
    if (tid < BC*OUTW) {
      int b = tid / OUTW, o = tid - b*OUTW;
      float acc = b_out[o];
      for (int j = 0; j < H; ++j) acc += sH[b*H + j] * W_out[j*OUTW + o];
      for (int d = 0; d < D; ++d) acc += sR[b*D + d] * W_out[(H + d)*OUTW + o];
      acc = clipf(acc);
      if (t >= tOutStart) {
        out[(size_t)(b0 + b) * TOUT * OUTW + (t - tOutStart) * OUTW + o] = sigm(acc);
      }
    }
    __syncthreads();
  }
}

extern "C" void kernel_launch(void* const* d_in, const int* in_sizes, int n_in,
                              void* d_out, int out_size, void* d_ws, size_t ws_size,
                              hipStream_t stream) {
  (void)in_sizes; (void)n_in; (void)out_size; (void)ws_size;
  const float* inputs   = (const float*)d_in[0];
  const float* Wx       = (const float*)d_in[1];
  const float* Wh       = (const float*)d_in[2];
  const float* b_lstm   = (const float*)d_in[3];
  const float* W_params = (const float*)d_in[4];
  const float* b_params = (const float*)d_in[5];
  const float* W_out    = (const float*)d_in[6];
  const float* b_out    = (const float*)d_in[7];
  const int*   seq      = (const int*)d_in[8];
  float* out = (float*)d_out;
  float* ws  = (float*)d_ws;

  // prologue: pack [Wx;Wh] -> 132x400, pad W_params/b_params -> 100x96 / 96
  pack_weights<<<dim3((WS_TOT + 255) / 256), dim3(256), 0, stream>>>(
      Wx, Wh, W_params, b_params, ws);

  const size_t shmem = (size_t)SMEM_FLOATS * sizeof(float);   // ~267 KB < 320 KB/WGP
  (void)hipFuncSetAttribute((const void*)ntm_fused,
                            hipFuncAttributeMaxDynamicSharedMemorySize,
                            (int)shmem);

  ntm_fused<<<dim3(B_TOT / BC), dim3(NT), shmem, stream>>>(
      inputs, ws + WS_WCAT, b_lstm, ws + WS_WP, ws + WS_BP,
      W_out, b_out, seq, out);
}